// PEPSLayer_55533927137912
// MI455X (gfx1250) — compile-verified
//
#include <hip/hip_runtime.h>

// PEPS 4x4 contraction, D=6, P=2, F=10, one workgroup (256 thr) per sample.
// Node L2 norms cancel into the final normalization -> skipped.
// Environment contraction order: peak intermediate = D^5 (padded 224x48 in LDS).
// Each tensordot: pack to padded canonical panels, then v_wmma_f32_16x16x4_f32.
// Pack = unconditional valid-region gather + constexpr-sized pad-zero loops.
// Node build vectorized to float4 (global_load_b128 -> ds_store_b128).

typedef float v2f __attribute__((ext_vector_type(2)));
typedef float v8f __attribute__((ext_vector_type(8)));

// ---- LDS layout (floats) ----
#define OFF_BUF 5760           // 10752: padded intermediate buffer (224 x 48)
#define OFF_TB  16512          // 1296: top boundary  T[v1_0..v1_3] (compact 216x6)
#define OFF_BB  17808          // 1296: bottom bound. B[v2_0..v2_3] (compact 216x6)
#define OFF_AP  19104          // 8064: packed A panel (224 x 36 max)
#define OFF_BP  27168          // 1728: packed B panel ( 48 x 36 max)
#define LDS_FLOATS 28896       // 115,584 bytes

struct NodeDesc { int off, widx, xoff, n; };
__constant__ NodeDesc g_nodes[15] = {
  {0,     0,  0,   36}, {36,    1,  2,  216}, {252,   2,  4,  216}, {468,   3,  6,   36},
  {504,   4,  8,  216}, {720,   5, 10, 1296}, {2016,  6, 12, 1296}, {3312,  7, 14,  216},
  {3528,  8, 16,  216}, {3744,  9, 18, 1296}, /* center w22 (widx 10) built on the fly */
  {5040, 11, 22,  216}, {5256, 12, 24,   36}, {5292, 13, 26,  216}, {5508, 14, 28,  216},
  {5724, 15, 30,   36},
};

struct WPtrs { const float* w[16]; };

// One contraction step: C = A x B over shared base-6 digit indices, LDS only.
// A addressing: m = m2*36+m1*6+m0 (MS2,MS1,MS0), k = k1*6+k0 (AKS1,AKS0).
// B addressing: k digits (BKS1,BKS0), n = n1*6+n0 (NS1,NS0).
// PADC: store C padded [Mp][Np] unconditionally; else compact [M][N] guarded.
template<int M, int K, int N,
         int MS2, int MS1, int MS0, int AKS1, int AKS0,
         int BKS1, int BKS0, int NS1, int NS0, bool PADC>
__device__ __forceinline__ void step(const float* A, const float* Bn, float* C,
                                     float* Ap, float* Bp, int tid) {
  constexpr int Mt = (M + 15) / 16, Nt = (N + 15) / 16, Kt = (K + 3) / 4;
  constexpr int Mp = Mt * 16, Np = Nt * 16, Kp = Kt * 4;

  // ---- pack A -> Ap[m*Kp + k]: unconditional valid-region gather ----
  for (int i = tid; i < M * K; i += 256) {
    int m = i / K, k = i - m * K;
    int m0 = m % 6, mq = m / 6, m1 = mq % 6, m2 = mq / 6;
    int k0 = k % 6, k1 = k / 6;
    Ap[m * Kp + k] = A[m2 * MS2 + m1 * MS1 + m0 * MS0 + k1 * AKS1 + k0 * AKS0];
  }
  if (Mp > M) {                       // zero pad rows [M, Mp)
    for (int i = tid; i < (Mp - M) * Kp; i += 256) Ap[M * Kp + i] = 0.f;
  }
  if (Kp > K) {                       // zero pad cols [K, Kp) of valid rows
    for (int i = tid; i < M * (Kp - K); i += 256) {
      int m = i / (Kp - K), k = K + (i - m * (Kp - K));
      Ap[m * Kp + k] = 0.f;
    }
  }
  // ---- pack B (transposed, K contiguous) -> Bp[n*Kp + k] ----
  for (int i = tid; i < N * K; i += 256) {
    int n = i / K, k = i - n * K;
    int k0 = k % 6, k1 = k / 6;
    int n0 = n % 6, n1 = n / 6;
    Bp[n * Kp + k] = Bn[k1 * BKS1 + k0 * BKS0 + n1 * NS1 + n0 * NS0];
  }
  if (Np > N) {
    for (int i = tid; i < (Np - N) * Kp; i += 256) Bp[N * Kp + i] = 0.f;
  }
  if (Kp > K) {
    for (int i = tid; i < N * (Kp - K); i += 256) {
      int n = i / (Kp - K), k = K + (i - n * (Kp - K));
      Bp[n * Kp + k] = 0.f;
    }
  }
  __syncthreads();

  // ---- WMMA tiles: no bounds checks, contiguous b64 fragment loads ----
  const int lane  = tid & 31;
  const int wave  = tid >> 5;
  const int lnlo  = lane & 15;
  const int khalf = (lane >> 4) << 1;   // A/B: lanes 16-31 hold K = {2,3}
  const int mhi   = (lane >> 4) << 3;   // C/D: lanes 16-31 hold rows r+8
  for (int t = wave; t < Mt * Nt; t += 8) {
    int mt = t / Nt, nt = t - mt * Nt;
    const float* Arow = Ap + (mt * 16 + lnlo) * Kp + khalf;
    const float* Brow = Bp + (nt * 16 + lnlo) * Kp + khalf;
    v8f acc = {0.f, 0.f, 0.f, 0.f, 0.f, 0.f, 0.f, 0.f};
    #pragma unroll
    for (int kb = 0; kb < Kt; ++kb) {
      v2f av = *(const v2f*)(Arow + kb * 4);   // ds_load_b64 (pairs -> 2addr)
      v2f bv = *(const v2f*)(Brow + kb * 4);
      acc = __builtin_amdgcn_wmma_f32_16x16x4_f32(
          false, av, false, bv, (short)0, acc, false, false);
    }
    int nn = nt * 16 + lnlo;
    #pragma unroll
    for (int r = 0; r < 8; ++r) {
      int mm = mt * 16 + mhi + r;
      if (PADC) {
        C[mm * Np + nn] = acc[r];              // unconditional padded store
      } else {
        if (mm < M && nn < N) C[mm * N + nn] = acc[r];
      }
    }
  }
  __syncthreads();
}

__global__ void __launch_bounds__(256)
peps_contract_kernel(const float* __restrict__ X, WPtrs W, float* __restrict__ out) {
  __shared__ float lds[LDS_FLOATS];
  const int b   = blockIdx.x;
  const int tid = threadIdx.x;
  const float* xb = X + b * 32;   // x[b,i,j,p] = xb[i*8 + j*2 + p]

  // ---- build unnormalized node tensors t = x0*w[0,..] + x1*w[1,..] ----
  // All node sizes and LDS offsets are multiples of 4 -> float4 path
  // (global_load_b128 + ds_store_b128).
  for (int nd = 0; nd < 15; ++nd) {
    NodeDesc d = g_nodes[nd];
    float x0 = xb[d.xoff], x1 = xb[d.xoff + 1];
    const float4* wa = (const float4*)W.w[d.widx];
    const float4* wb = (const float4*)(W.w[d.widx] + d.n);
    float4* dst = (float4*)(lds + d.off);
    int n4 = d.n >> 2;
    for (int i = tid; i < n4; i += 256) {
      float4 a = wa[i], c = wb[i], r;
      r.x = x0 * a.x + x1 * c.x;
      r.y = x0 * a.y + x1 * c.y;
      r.z = x0 * a.z + x1 * c.z;
      r.w = x0 * a.w + x1 * c.w;
      dst[i] = r;
    }
  }
  __syncthreads();

  float* BUF = lds + OFF_BUF;
  float* TB  = lds + OFF_TB;
  float* BB  = lds + OFF_BB;
  float* Ap  = lds + OFF_AP;
  float* Bp  = lds + OFF_BP;

  // ---- 14 contraction steps; stride tables account for padded producers ----
  // S1: n00[v0_0,h0_0] x n01 -> padded [v0_0 | v0_1,h0_1], pitch 48
  step<  6, 6,36,    0,   0,  6,    0,  1,     0, 36,  6, 1, true >(lds + 0,    lds + 36,   BUF, Ap, Bp, tid);
  // S2: x n02 -> [v0_0,v0_1 | v0_2,h0_2] (A strides from pitch-48 C1)
  step< 36, 6,36,    0,  48,  6,    0,  1,     0, 36,  6, 1, true >(BUF,        lds + 252,  BUF, Ap, Bp, tid);
  // S3: x n03 -> T0 [v0_0,v0_1,v0_2 | v0_3], pitch 16
  step<216, 6, 6,  288,  48,  6,    0,  1,     0,  6,  0, 1, true >(BUF,        lds + 468,  BUF, Ap, Bp, tid);
  // S4: T0 x n10 (shared v0_0 leading, pitch-16 T0) -> [v0_1,v0_2,v0_3 | v1_0,h1_0]
  step<216, 6,36,   96,  16,  1,    0,576,     0, 36,  6, 1, true >(BUF,        lds + 504,  BUF, Ap, Bp, tid);
  // S5: x n11 (shared v0_1,h1_0) -> [v0_2,v0_3,v1_0 | v1_1,h1_1]
  step<216,36,36,  288,  48,  6, 1728,  1,   216, 36,  6, 1, true >(BUF,        lds + 720,  BUF, Ap, Bp, tid);
  // S6: x n12 -> [v0_3,v1_0,v1_1 | v1_2,h1_2]
  step<216,36,36,  288,  48,  6, 1728,  1,   216, 36,  6, 1, true >(BUF,        lds + 2016, BUF, Ap, Bp, tid);
  // S7: x n13 -> T compact [216][6] = [v1_0,v1_1,v1_2,v1_3]
  step<216,36, 6,  288,  48,  6, 1728,  1,    36,  6,  0, 1, false>(BUF,        lds + 3312, TB,  Ap, Bp, tid);
  // S8: n30[v2_0,h3_0] x n31 -> [v2_0 | v2_1,h3_1], pitch 48
  step<  6, 6,36,    0,   0,  6,    0,  1,     0,  6, 36, 1, true >(lds + 5256, lds + 5292, BUF, Ap, Bp, tid);
  // S9: x n32 -> [v2_0,v2_1 | v2_2,h3_2]
  step< 36, 6,36,    0,  48,  6,    0,  1,     0,  6, 36, 1, true >(BUF,        lds + 5508, BUF, Ap, Bp, tid);
  // S10: x n33 (shared h3_2, free v2_3 via B^T) -> Bt compact [216][6]
  step<216, 6, 6,  288,  48,  6,    0,  1,     0,  1,  0, 6, false>(BUF,        lds + 5724, BB,  Ap, Bp, tid);
  // S11: T (compact) x n20 -> [v1_1,v1_2,v1_3 | v2_0,h2_0]
  step<216, 6,36,   36,   6,  1,    0,216,     0, 36,  6, 1, true >(TB,         lds + 3528, BUF, Ap, Bp, tid);
  // S12: x n21 (shared v1_1,h2_0) -> [v1_2,v1_3,v2_0 | v2_1,h2_1]
  step<216,36,36,  288,  48,  6, 1728,  1,   216, 36,  6, 1, true >(BUF,        lds + 3744, BUF, Ap, Bp, tid);
  // S13: x Bt (shared v2_0,v2_1; A mid-block strides 48,6) -> [v1_2,v1_3,h2_1 | v2_2,v2_3]
  step<216,36,36, 1728, 288,  1,   48,  6,   216, 36,  6, 1, true >(BUF,        BB,         BUF, Ap, Bp, tid);
  // S14: x n23 (shared v1_3,v2_3) -> env [v1_2,h2_1,v2_2 | h2_2], pitch 16
  step<216,36, 6, 1728,  48,  6,  288,  1,    36,  1,  0, 6, true >(BUF,        lds + 5040, BUF, Ap, Bp, tid);

  // ---- final: out[o] = env . center(:,o), center built on the fly, normalize ----
  {
    float x0 = xb[20], x1 = xb[21];          // x[b,2,2,:]
    const float* w22 = W.w[10];              // [2,6,6,6,6,10]
    const float* env = BUF;                  // padded: elem e -> env[(e/6)*16 + e%6]
    float accv[10];
    #pragma unroll
    for (int o = 0; o < 10; ++o) accv[o] = 0.f;
    for (int e = tid; e < 1296; e += 256) {
      float ev = env[(e / 6) * 16 + (e % 6)];
      const float* wa = w22 + e * 10;
      const float* wb = wa + 12960;
      #pragma unroll
      for (int o = 0; o < 10; ++o)
        accv[o] += ev * (x0 * wa[o] + x1 * wb[o]);
    }
    float* red = Ap;                         // 2570 floats of scratch fits in Ap
    #pragma unroll
    for (int o = 0; o < 10; ++o) red[o * 256 + tid] = accv[o];
    __syncthreads();
    if (tid < 10) {
      float s = 0.f;
      for (int i = 0; i < 256; ++i) s += red[tid * 256 + i];
      red[2560 + tid] = s;
    }
    __syncthreads();
    if (tid == 0) {
      float nrm = 0.f;
      #pragma unroll
      for (int o = 0; o < 10; ++o) { float v = red[2560 + o]; nrm += v * v; }
      nrm = sqrtf(nrm);
      float inv = (nrm > 0.f) ? 1.f / nrm : 0.f;
      #pragma unroll
      for (int o = 0; o < 10; ++o) out[b * 10 + o] = red[2560 + o] * inv;
    }
  }
}

extern "C" void kernel_launch(void* const* d_in, const int* in_sizes, int n_in,
                              void* d_out, int out_size, void* d_ws, size_t ws_size,
                              hipStream_t stream) {
  (void)n_in; (void)out_size; (void)d_ws; (void)ws_size;
  const float* X = (const float*)d_in[0];
  WPtrs W;
  for (int i = 0; i < 16; ++i) W.w[i] = (const float*)d_in[1 + i];
  int B = in_sizes[0] / 32;                  // x is [B,4,4,2]
  peps_contract_kernel<<<B, 256, 0, stream>>>(X, W, (float*)d_out);
}